// EventSpecificTimingHeads_57681410786058
// MI455X (gfx1250) — compile-verified
//
#include <hip/hip_runtime.h>

typedef float v2f __attribute__((ext_vector_type(2)));
typedef float v8f __attribute__((ext_vector_type(8)));

#define WMMA_F32(a, b, c) \
  __builtin_amdgcn_wmma_f32_16x16x4_f32(false, (a), false, (b), (short)0, (c), false, false)

static __device__ __forceinline__ v2f ld2(const float* p) { return *(const v2f*)p; }

// Problem constants
#define EV   16
#define BB   8
#define SS   512
#define DD   128
#define HH   4
#define DH   32
#define H2   64
#define N3   384      // 3*D
#define BS   4096     // B*S

// ---------------------------------------------------------------------------
// Kernel 1: qkv[e, m, n] = sum_d X[m, d] * Wqkv[e, n, d] + bqkv[e, n]
// One wave per 16x16 output tile, K=128 -> 32 x v_wmma_f32_16x16x4_f32.
// ---------------------------------------------------------------------------
__global__ __launch_bounds__(32) void k_qkv(const float* __restrict__ X,
                                            const float* __restrict__ Wq,
                                            const float* __restrict__ bq,
                                            float* __restrict__ qkv) {
  int bid = blockIdx.x;
  int nt  = bid % 24;            // 384 / 16
  int mt  = (bid / 24) & 255;    // 4096 / 16
  int e   = bid / (24 * 256);
  int lane = threadIdx.x;
  int half = lane >> 4, lr = lane & 15;

  const float* Arow = X  + (size_t)(mt * 16 + lr) * DD + half * 2;
  const float* Brow = Wq + ((size_t)e * N3 + nt * 16 + lr) * DD + half * 2;

  v8f acc = {0.f, 0.f, 0.f, 0.f, 0.f, 0.f, 0.f, 0.f};
#pragma unroll
  for (int kk = 0; kk < DD; kk += 4) {
    v2f a = ld2(Arow + kk);
    v2f b = ld2(Brow + kk);
    acc = WMMA_F32(a, b, acc);
  }

  float bias = bq[e * N3 + nt * 16 + lr];
  float* outp = qkv + ((size_t)e * BS + mt * 16) * N3 + nt * 16 + lr;
#pragma unroll
  for (int v = 0; v < 8; ++v)
    outp[(size_t)(v + half * 8) * N3] = acc[v] + bias;
}

// ---------------------------------------------------------------------------
// Kernel 2: attention per (e, b, h, q-tile of 16 rows).
// scores -> LDS (padded), softmax, ctx = P @ V, all via f32 WMMA.
// ---------------------------------------------------------------------------
#define SROW 516   // 512 + 4 pad: kills LDS bank conflicts (516 % 64 == 4)

__global__ __launch_bounds__(32) void k_attn(const float* __restrict__ qkv,
                                             float* __restrict__ ctx) {
  __shared__ float sc[16 * SROW];
  __shared__ float rinv[16];

  int bid = blockIdx.x;
  int qt = bid & 31;
  int h  = (bid >> 5) & 3;
  int b  = (bid >> 7) & 7;
  int e  = bid >> 10;
  int lane = threadIdx.x, half = lane >> 4, lr = lane & 15;

  size_t rowBase = ((size_t)e * BS + (size_t)b * SS) * N3;

  // Preload q fragments for this q-tile (reused across all 32 k-tiles)
  v2f qf[8];
  const float* qp = qkv + rowBase + (size_t)(qt * 16 + lr) * N3 + h * DH + half * 2;
#pragma unroll
  for (int i = 0; i < 8; ++i) qf[i] = ld2(qp + i * 4);

  const float scale = 0.17677669529663687f;  // 1/sqrt(32)

  // scores = scale * q k^T  -> LDS (unnormalized)
  for (int kt = 0; kt < 32; ++kt) {
    const float* kp = qkv + rowBase + (size_t)(kt * 16 + lr) * N3 + DD + h * DH + half * 2;
    v8f acc = {0.f, 0.f, 0.f, 0.f, 0.f, 0.f, 0.f, 0.f};
#pragma unroll
    for (int i = 0; i < 8; ++i) {
      v2f bf = ld2(kp + i * 4);
      acc = WMMA_F32(qf[i], bf, acc);
    }
#pragma unroll
    for (int v = 0; v < 8; ++v)
      sc[(v + half * 8) * SROW + kt * 16 + lr] = acc[v] * scale;
  }
  __syncthreads();

  // softmax: lane handles row lr, one half (256 cols); combine halves via shfl_xor(16)
  {
    float* rp = sc + lr * SROW + half * 256;
    float m = -3.4e38f;
    for (int j = 0; j < 256; ++j) m = fmaxf(m, rp[j]);
    m = fmaxf(m, __shfl_xor(m, 16, 32));
    float ssum = 0.f;
    for (int j = 0; j < 256; ++j) {
      float ev = __expf(rp[j] - m);
      rp[j] = ev;
      ssum += ev;
    }
    ssum += __shfl_xor(ssum, 16, 32);
    if (half == 0) rinv[lr] = 1.0f / ssum;
  }
  __syncthreads();

  // ctx = P @ V : 16 x 32 output (two 16x16 n-tiles), K = 512
  v8f c0 = {0.f, 0.f, 0.f, 0.f, 0.f, 0.f, 0.f, 0.f};
  v8f c1 = {0.f, 0.f, 0.f, 0.f, 0.f, 0.f, 0.f, 0.f};
  const float* vbase = qkv + rowBase + 2 * DD + h * DH;
#pragma unroll 4
  for (int kk = 0; kk < SS; kk += 4) {
    v2f a = ld2(sc + lr * SROW + kk + half * 2);   // P fragment from LDS
    int krow = kk + half * 2;
    const float* vp = vbase + (size_t)krow * N3;
    v2f b0, b1;
    b0.x = vp[lr];           b0.y = vp[N3 + lr];
    b1.x = vp[16 + lr];      b1.y = vp[N3 + 16 + lr];
    c0 = WMMA_F32(a, b0, c0);
    c1 = WMMA_F32(a, b1, c1);
  }

  float* cp = ctx + ((size_t)e * BS + (size_t)b * SS + qt * 16) * DD + h * DH;
#pragma unroll
  for (int v = 0; v < 8; ++v) {
    int r = v + half * 8;
    float inv = rinv[r];
    cp[(size_t)r * DD + lr]      = c0[v] * inv;
    cp[(size_t)r * DD + 16 + lr] = c1[v] * inv;
  }
}

// ---------------------------------------------------------------------------
// Kernel 3: fused out-proj + bias, ReLU MLP (128->64), head (64->1).
// One wave per (e, 16-row tile of B*S).
// ---------------------------------------------------------------------------
#define AST 132   // 128 + 4 pad
#define HST 68    // 64 + 4 pad

__global__ __launch_bounds__(32) void k_head(const float* __restrict__ ctx,
                                             const float* __restrict__ Wo,
                                             const float* __restrict__ bo,
                                             const float* __restrict__ W1,
                                             const float* __restrict__ b1,
                                             const float* __restrict__ W2,
                                             const float* __restrict__ b2,
                                             float* __restrict__ out) {
  __shared__ float att[16 * AST];
  __shared__ float hh[16 * HST];

  int bid = blockIdx.x;
  int mt = bid & 255;
  int e  = bid >> 8;
  int lane = threadIdx.x, half = lane >> 4, lr = lane & 15;

  // Preload ctx A-fragments for the 16-row tile (reused across all 8 n-tiles)
  v2f af[32];
  const float* ap = ctx + ((size_t)e * BS + mt * 16 + lr) * DD + half * 2;
#pragma unroll
  for (int i = 0; i < 32; ++i) af[i] = ld2(ap + i * 4);

  // attended = ctx @ Wo^T + bo   (16 x 128)
  for (int nt = 0; nt < 8; ++nt) {
    const float* wp = Wo + ((size_t)e * DD + nt * 16 + lr) * DD + half * 2;
    v8f acc = {0.f, 0.f, 0.f, 0.f, 0.f, 0.f, 0.f, 0.f};
#pragma unroll
    for (int i = 0; i < 32; ++i) acc = WMMA_F32(af[i], ld2(wp + i * 4), acc);
    float bias = bo[e * DD + nt * 16 + lr];
#pragma unroll
    for (int v = 0; v < 8; ++v)
      att[(v + half * 8) * AST + nt * 16 + lr] = acc[v] + bias;
  }
  __syncthreads();

  // h = relu(attended @ W1^T + b1)   (16 x 64)
  for (int nt = 0; nt < 4; ++nt) {
    const float* wp = W1 + ((size_t)e * H2 + nt * 16 + lr) * DD + half * 2;
    v8f acc = {0.f, 0.f, 0.f, 0.f, 0.f, 0.f, 0.f, 0.f};
#pragma unroll
    for (int i = 0; i < 32; ++i) {
      v2f a = ld2(att + lr * AST + i * 4 + half * 2);
      acc = WMMA_F32(a, ld2(wp + i * 4), acc);
    }
    float bias = b1[e * H2 + nt * 16 + lr];
#pragma unroll
    for (int v = 0; v < 8; ++v)
      hh[(v + half * 8) * HST + nt * 16 + lr] = fmaxf(acc[v] + bias, 0.f);
  }
  __syncthreads();

  // logits = h @ W2^T + b2 -> out[b, s, e]  (lane: row lr, half of 64 dims)
  float acc3 = 0.f;
  const float* w2 = W2 + e * H2 + half * 32;
  const float* hp = hh + lr * HST + half * 32;
#pragma unroll
  for (int j = 0; j < 32; ++j) acc3 += hp[j] * w2[j];
  acc3 += __shfl_xor(acc3, 16, 32);
  if (half == 0)
    out[(size_t)(mt * 16 + lr) * EV + e] = acc3 + b2[e];
}

// ---------------------------------------------------------------------------
extern "C" void kernel_launch(void* const* d_in, const int* in_sizes, int n_in,
                              void* d_out, int out_size, void* d_ws, size_t ws_size,
                              hipStream_t stream) {
  (void)in_sizes; (void)n_in; (void)out_size; (void)ws_size;
  const float* X    = (const float*)d_in[0];
  const float* Wqkv = (const float*)d_in[1];
  const float* bqkv = (const float*)d_in[2];
  const float* Wo   = (const float*)d_in[3];
  const float* bo   = (const float*)d_in[4];
  const float* W1   = (const float*)d_in[5];
  const float* b1   = (const float*)d_in[6];
  const float* W2   = (const float*)d_in[7];
  const float* b2   = (const float*)d_in[8];

  float* qkv = (float*)d_ws;                         // 16*4096*384 f32 = 96 MB
  float* ctx = qkv + (size_t)EV * BS * N3;           // 16*4096*128 f32 = 32 MB

  k_qkv <<<dim3(EV * 256 * 24), dim3(32), 0, stream>>>(X, Wqkv, bqkv, qkv);
  k_attn<<<dim3(EV * BB * HH * 32), dim3(32), 0, stream>>>(qkv, ctx);
  k_head<<<dim3(EV * 256), dim3(32), 0, stream>>>(ctx, Wo, bo, W1, b1, W2, b2,
                                                  (float*)d_out);
}